// ProsodyPredictorV15_63213328662960
// MI455X (gfx1250) — compile-verified
//
#include <hip/hip_runtime.h>
#include <hip/hip_bf16.h>

// Problem constants (from reference)
#define Bv      16
#define Tv      1024
#define MELv    80
#define Cv      512
#define NLv     6
#define NSTATEv 32
#define NFREQv  513
#define NCH     8            // time chunks for the parallel scan
#define CHUNK   (Tv / NCH)   // 128 steps per chunk
#define STAGE   16           // timesteps per async-LDS staging batch
#define NSTG    (CHUNK / STAGE)

typedef float v2f __attribute__((ext_vector_type(2)));
typedef float v8f __attribute__((ext_vector_type(8)));

// ---------------------------------------------------------------------------
// abar[l,c,n] = exp(-exp(logA[l,c,n]))   (per-step decay of the diagonal SSM)
// ---------------------------------------------------------------------------
__global__ void build_abar(const float* __restrict__ logA, float* __restrict__ abar) {
    int i = blockIdx.x * 256 + threadIdx.x;
    if (i < NLv * Cv * NSTATEv) abar[i] = expf(-expf(logA[i]));
}

// ---------------------------------------------------------------------------
// h0 = mel @ w_in + b_in + freq   via V_WMMA_F32_16X16X4_F32
// One wave computes one 16x16 tile of (B*T, C); K=80 -> 20 wmma steps.
// Layouts per CDNA5 ISA 7.12.2:
//   A 16x4 : lane m=lane&15, VGPR pair holds K = 2*(lane>>4) + {0,1}
//   B 4x16 : lane n=lane&15, VGPR pair holds K = 2*(lane>>4) + {0,1}
//   D 16x16: VGPR r -> M = r + 8*(lane>>4), N = lane&15
// ---------------------------------------------------------------------------
__global__ void __launch_bounds__(32)
proj_wmma(const float* __restrict__ mel, const float* __restrict__ w_in,
          const float* __restrict__ b_in, const float* __restrict__ freq,
          float* __restrict__ h0) {
    const int n0   = blockIdx.x * 16;
    const int m0   = blockIdx.y * 16;
    const int lane = threadIdx.x;
    const int mr   = lane & 15;
    const int half = lane >> 4;
    const int kb   = half * 2;

    v8f acc = {};
    for (int k0 = 0; k0 < MELv; k0 += 4) {
        v2f av, bv;
        const float* ap = mel + (size_t)(m0 + mr) * MELv + (k0 + kb);
        av.x = ap[0];
        av.y = ap[1];
        bv.x = w_in[(size_t)(k0 + kb)     * Cv + n0 + mr];
        bv.y = w_in[(size_t)(k0 + kb + 1) * Cv + n0 + mr];
        acc = __builtin_amdgcn_wmma_f32_16x16x4_f32(
            /*neg_a=*/false, av, /*neg_b=*/false, bv,
            /*c_mod=*/(short)0, acc, /*reuse_a=*/false, /*reuse_b=*/false);
    }

    const int   col = n0 + mr;
    const float bb  = b_in[col];
#pragma unroll
    for (int r = 0; r < 8; ++r) {
        int row = m0 + r + 8 * half;          // row in [0, B*T)
        int t   = row & (Tv - 1);             // T tiles never straddle batches
        int f   = t < NFREQv ? t : (NFREQv - 1); // pad: broadcast last freq row
        h0[(size_t)row * Cv + col] = acc[r] + bb + freq[(size_t)f * Cv + col];
    }
}

// ---------------------------------------------------------------------------
// Issue one async staging batch: STAGE timesteps of h (this lane's channel)
// into LDS. One GLOBAL_LOAD_ASYNC_TO_LDS_B32 moves 32 lanes x 4B per step;
// completion is tracked on ASYNCcnt (in-order for async loads).
// ---------------------------------------------------------------------------
__device__ __forceinline__ void stage_issue(const float* gp, unsigned lbase) {
#pragma unroll
    for (int k = 0; k < STAGE; ++k) {
        asm volatile("global_load_async_to_lds_b32 %0, %1, off"
                     :: "v"(lbase + (unsigned)(k * 128)),   // LDS byte addr (32 lanes * 4B rows)
                        "v"(gp + (size_t)k * Cv)            // per-lane 64-bit global addr
                     : "memory");
    }
}

// ---------------------------------------------------------------------------
// Scan phase 1: per-chunk local state from zero init.
// S_local[n] = sum_{i<CHUNK} a_n^(CHUNK-1-i) * h[t0+i]
// One wave per (b, 32-channel group, chunk); 32 states in registers per lane.
// h is streamed through LDS with double-buffered async loads so the global
// latency never touches the dependent FMA chain.
// ---------------------------------------------------------------------------
__global__ void __launch_bounds__(32)
chunk_state(const float* __restrict__ hin, const float* __restrict__ abar_l,
            float* __restrict__ S) {
    const int b = blockIdx.x, cg = blockIdx.y, j = blockIdx.z;
    const int lane = threadIdx.x;
    const int c = cg * 32 + lane;

    __shared__ float sbuf[2][STAGE][32];

    float a[NSTATEv], z[NSTATEv];
#pragma unroll
    for (int n = 0; n < NSTATEv; ++n) { a[n] = abar_l[(size_t)c * NSTATEv + n]; z[n] = 0.f; }

    const float* hp = hin + ((size_t)b * Tv + (size_t)j * CHUNK) * Cv + c;
    const unsigned lds0 = (unsigned)(unsigned long long)&sbuf[0][0][lane];
    const unsigned lds1 = (unsigned)(unsigned long long)&sbuf[1][0][lane];

    stage_issue(hp, lds0);
    stage_issue(hp + (size_t)STAGE * Cv, lds1);

    for (int s = 0; s < NSTG; ++s) {
        if (s + 1 < NSTG) asm volatile("s_wait_asynccnt 0x10" ::: "memory"); // stage s done
        else              asm volatile("s_wait_asynccnt 0x0"  ::: "memory");
        const float* lp = &sbuf[s & 1][0][lane];
#pragma unroll
        for (int k = 0; k < STAGE; ++k) {
            const float hv = lp[(size_t)k * 32];
#pragma unroll
            for (int n = 0; n < NSTATEv; ++n) z[n] = __builtin_fmaf(a[n], z[n], hv);
        }
        if (s + 2 < NSTG) stage_issue(hp + (size_t)(s + 2) * STAGE * Cv,
                                      (s & 1) ? lds1 : lds0);
    }

    float* sp = S + (((size_t)j * Bv + b) * Cv + c) * NSTATEv;
#pragma unroll
    for (int n = 0; n < NSTATEv; ++n) sp[n] = z[n];
}

// ---------------------------------------------------------------------------
// Scan phase 2: cross-chunk scan (8 steps). Overwrites S[j] with the INCOMING
// state of chunk j:  X_in(j+1) = a^CHUNK * X_in(j) + S_local(j),  X_in(0)=0.
// One thread per (b,c,n).
// ---------------------------------------------------------------------------
__global__ void chunk_scan(const float* __restrict__ logA_l, float* __restrict__ S) {
    const int idx = blockIdx.x * 256 + threadIdx.x;          // B*C*NSTATE threads
    const int cn  = idx & (Cv * NSTATEv - 1);
    const float A    = -expf(logA_l[cn]);
    const float aChk = expf((float)CHUNK * A);               // a^CHUNK
    const int stride = Bv * Cv * NSTATEv;

    float X = 0.f;
    for (int j = 0; j < NCH; ++j) {
        const float loc = S[(size_t)j * stride + idx];
        S[(size_t)j * stride + idx] = X;
        X = __builtin_fmaf(aChk, X, loc);
    }
}

// ---------------------------------------------------------------------------
// Scan phase 3: replay each chunk from its incoming state, fused with
// y = sum_n C_n x_n,  out = gelu(y + D*h)  (tanh-approx gelu, as jax.nn.gelu)
// Same async-LDS double-buffered streaming of h as phase 1.
// ---------------------------------------------------------------------------
__global__ void __launch_bounds__(32)
chunk_out(const float* __restrict__ hin, const float* __restrict__ abar_l,
          const float* __restrict__ Cp_l, const float* __restrict__ D_l,
          const float* __restrict__ S, float* __restrict__ hout) {
    const int b = blockIdx.x, cg = blockIdx.y, j = blockIdx.z;
    const int lane = threadIdx.x;
    const int c = cg * 32 + lane;

    __shared__ float sbuf[2][STAGE][32];

    float a[NSTATEv], cp[NSTATEv], z[NSTATEv];
    const float* sp = S + (((size_t)j * Bv + b) * Cv + c) * NSTATEv;
#pragma unroll
    for (int n = 0; n < NSTATEv; ++n) {
        a[n]  = abar_l[(size_t)c * NSTATEv + n];
        cp[n] = Cp_l[(size_t)c * NSTATEv + n];
        z[n]  = sp[n];
    }
    const float dv = D_l[c];

    const size_t base = ((size_t)b * Tv + (size_t)j * CHUNK) * Cv + c;
    const float* hp = hin + base;
    float*       op = hout + base;

    const unsigned lds0 = (unsigned)(unsigned long long)&sbuf[0][0][lane];
    const unsigned lds1 = (unsigned)(unsigned long long)&sbuf[1][0][lane];

    stage_issue(hp, lds0);
    stage_issue(hp + (size_t)STAGE * Cv, lds1);

    for (int s = 0; s < NSTG; ++s) {
        if (s + 1 < NSTG) asm volatile("s_wait_asynccnt 0x10" ::: "memory");
        else              asm volatile("s_wait_asynccnt 0x0"  ::: "memory");
        const float* lp = &sbuf[s & 1][0][lane];
#pragma unroll
        for (int k = 0; k < STAGE; ++k) {
            const float hv = lp[(size_t)k * 32];
            float y = 0.f;
#pragma unroll
            for (int n = 0; n < NSTATEv; ++n) {
                z[n] = __builtin_fmaf(a[n], z[n], hv);
                y    = __builtin_fmaf(cp[n], z[n], y);
            }
            const float u   = __builtin_fmaf(dv, hv, y);
            const float tin = 0.7978845608028654f * (u + 0.044715f * u * u * u);
            op[(size_t)(s * STAGE + k) * Cv] = 0.5f * u * (1.f + tanhf(tin));
        }
        if (s + 2 < NSTG) stage_issue(hp + (size_t)(s + 2) * STAGE * Cv,
                                      (s & 1) ? lds1 : lds0);
    }
}

// ---------------------------------------------------------------------------
// hm[b,c] = mean over t of h[b,t,c]
// ---------------------------------------------------------------------------
__global__ void mean_time(const float* __restrict__ h, float* __restrict__ hm) {
    const int b = blockIdx.x;
    const int c = blockIdx.y * 256 + threadIdx.x;
    const float* p = h + (size_t)b * Tv * Cv + c;
    float s = 0.f;
    for (int t = 0; t < Tv; ++t) s += p[(size_t)t * Cv];
    hm[(size_t)b * Cv + c] = s * (1.f / Tv);
}

// ---------------------------------------------------------------------------
// LDS block reduction (blockDim.x == 256)
// ---------------------------------------------------------------------------
__device__ __forceinline__ float block_reduce_sum(float v, float* sdata) {
    const int tid = threadIdx.x;
    sdata[tid] = v;
    __syncthreads();
    for (int s = 128; s > 0; s >>= 1) {
        if (tid < s) sdata[tid] += sdata[tid + s];
        __syncthreads();
    }
    const float r = sdata[0];
    __syncthreads();
    return r;
}

// ---------------------------------------------------------------------------
// Per-time heads: f0 / energy / pitch_var. One block per (b,t).
// mean/var computed once; reused for the three (g,b) affine heads.
// ---------------------------------------------------------------------------
__global__ void time_heads(const float* __restrict__ h,
                           const float* __restrict__ ln_g, const float* __restrict__ ln_b,
                           const float* __restrict__ w5,  const float* __restrict__ b5,
                           float* __restrict__ f0, float* __restrict__ energy,
                           float* __restrict__ pv) {
    __shared__ float sdata[256];
    const int bt  = blockIdx.x;
    const int tid = threadIdx.x;
    const float* x = h + (size_t)bt * Cv;

    const float x0 = x[tid], x1 = x[tid + 256];
    const float mu = block_reduce_sum(x0 + x1, sdata) * (1.f / Cv);
    const float d0 = x0 - mu, d1 = x1 - mu;
    const float var  = block_reduce_sum(d0 * d0 + d1 * d1, sdata) * (1.f / Cv);
    const float rstd = rsqrtf(var + 1e-5f);

    float* outs[3] = {f0, energy, pv};
    for (int k = 0; k < 3; ++k) {
        const float p =
            (d0 * rstd * ln_g[k * Cv + tid]       + ln_b[k * Cv + tid])       * w5[k * Cv + tid] +
            (d1 * rstd * ln_g[k * Cv + tid + 256] + ln_b[k * Cv + tid + 256]) * w5[k * Cv + tid + 256];
        const float r = block_reduce_sum(p, sdata);
        if (tid == 0) outs[k][bt] = r + b5[k];
    }
}

// ---------------------------------------------------------------------------
// Utterance heads: speech_rate, pause_dur, mfcc. One block per batch b.
// ---------------------------------------------------------------------------
__global__ void utt_heads(const float* __restrict__ hm,
                          const float* __restrict__ ln_g, const float* __restrict__ ln_b,
                          const float* __restrict__ w5,  const float* __restrict__ b5,
                          const float* __restrict__ wm,  const float* __restrict__ bm,
                          float* __restrict__ sr, float* __restrict__ pd,
                          float* __restrict__ mfcc) {
    __shared__ float sdata[256];
    const int b   = blockIdx.x;
    const int tid = threadIdx.x;
    const float* x = hm + (size_t)b * Cv;

    const float x0 = x[tid], x1 = x[tid + 256];
    const float mu = block_reduce_sum(x0 + x1, sdata) * (1.f / Cv);
    const float d0 = x0 - mu, d1 = x1 - mu;
    const float var  = block_reduce_sum(d0 * d0 + d1 * d1, sdata) * (1.f / Cv);
    const float rstd = rsqrtf(var + 1e-5f);

    // speech_rate (ln 3, w5[3]) and pause_dur (ln 4, w5[4])
    float* souts[2] = {sr, pd};
    for (int k = 3; k <= 4; ++k) {
        const float p =
            (d0 * rstd * ln_g[k * Cv + tid]       + ln_b[k * Cv + tid])       * w5[k * Cv + tid] +
            (d1 * rstd * ln_g[k * Cv + tid + 256] + ln_b[k * Cv + tid + 256]) * w5[k * Cv + tid + 256];
        const float r = block_reduce_sum(p, sdata);
        if (tid == 0) souts[k - 3][b] = r + b5[k];
    }

    // mfcc (ln 5, wm: (C,13))
    const float xi0 = d0 * rstd * ln_g[5 * Cv + tid]       + ln_b[5 * Cv + tid];
    const float xi1 = d1 * rstd * ln_g[5 * Cv + tid + 256] + ln_b[5 * Cv + tid + 256];
    for (int jj = 0; jj < 13; ++jj) {
        const float p = xi0 * wm[(size_t)tid * 13 + jj] +
                        xi1 * wm[(size_t)(tid + 256) * 13 + jj];
        const float r = block_reduce_sum(p, sdata);
        if (tid == 0) mfcc[(size_t)b * 13 + jj] = r + bm[jj];
    }
}

// ---------------------------------------------------------------------------
extern "C" void kernel_launch(void* const* d_in, const int* in_sizes, int n_in,
                              void* d_out, int out_size, void* d_ws, size_t ws_size,
                              hipStream_t stream) {
    const float* mel   = (const float*)d_in[0];
    const float* w_in  = (const float*)d_in[1];
    const float* b_in  = (const float*)d_in[2];
    const float* freq  = (const float*)d_in[3];
    const float* logA  = (const float*)d_in[4];
    const float* s4C   = (const float*)d_in[5];
    const float* s4D   = (const float*)d_in[6];
    const float* ln_g  = (const float*)d_in[7];
    const float* ln_b  = (const float*)d_in[8];
    const float* w5    = (const float*)d_in[9];
    const float* b5    = (const float*)d_in[10];
    const float* wm    = (const float*)d_in[11];
    const float* bm    = (const float*)d_in[12];

    float* ws   = (float*)d_ws;
    const size_t HSZ = (size_t)Bv * Tv * Cv;          // 8,388,608 floats
    float* h0   = ws;
    float* h1   = h0 + HSZ;
    float* abar = h1 + HSZ;                            // NL*C*NSTATE
    float* S    = abar + (size_t)NLv * Cv * NSTATEv;   // NCH*B*C*NSTATE
    float* hm   = S + (size_t)NCH * Bv * Cv * NSTATEv; // B*C

    float* out    = (float*)d_out;
    float* f0     = out;
    float* energy = out + (size_t)Bv * Tv;
    float* pv     = out + (size_t)2 * Bv * Tv;
    float* sr     = out + (size_t)3 * Bv * Tv;
    float* pd     = sr + Bv;
    float* mfcc   = pd + Bv;

    build_abar<<<(NLv * Cv * NSTATEv + 255) / 256, 256, 0, stream>>>(logA, abar);
    proj_wmma<<<dim3(Cv / 16, (Bv * Tv) / 16), 32, 0, stream>>>(mel, w_in, b_in, freq, h0);

    float* cur = h0;
    float* nxt = h1;
    for (int l = 0; l < NLv; ++l) {
        const float* abl = abar + (size_t)l * Cv * NSTATEv;
        const float* lal = logA + (size_t)l * Cv * NSTATEv;
        const float* cpl = s4C  + (size_t)l * Cv * NSTATEv;
        const float* dl  = s4D  + (size_t)l * Cv;
        chunk_state<<<dim3(Bv, Cv / 32, NCH), 32, 0, stream>>>(cur, abl, S);
        chunk_scan<<<(Bv * Cv * NSTATEv) / 256, 256, 0, stream>>>(lal, S);
        chunk_out<<<dim3(Bv, Cv / 32, NCH), 32, 0, stream>>>(cur, abl, cpl, dl, S, nxt);
        float* t = cur; cur = nxt; nxt = t;
    }
    // 6 layers -> result back in h0 (== cur)

    mean_time<<<dim3(Bv, Cv / 256), 256, 0, stream>>>(cur, hm);
    time_heads<<<Bv * Tv, 256, 0, stream>>>(cur, ln_g, ln_b, w5, b5, f0, energy, pv);
    utt_heads<<<Bv, 256, 0, stream>>>(hm, ln_g, ln_b, w5, b5, wm, bm, sr, pd, mfcc);
}